// DispNetC_53996328846165
// MI455X (gfx1250) — compile-verified
//
#include <hip/hip_runtime.h>
#include <stdint.h>

typedef __attribute__((ext_vector_type(2))) float v2f;
typedef __attribute__((ext_vector_type(8))) float v8f;

#define C_TOT   256
#define HH      64
#define WW      128
#define D_MAX   40
#define CHUNK   16    // channels per K-chunk (double buffered)
#define NCHUNK  (C_TOT / CHUNK)          // 16
#define LPITCH  136   // chunk row pitch (dwords): 2*136 = 16 mod 64 -> conflict-free
#define OPITCH  129   // output staging pitch (dwords): conflict-free diagonal scatter
#define HW      (HH*WW)                  // 8192
#define CHW     (C_TOT*HH*WW)            // 2097152

// Async DMA: global -> LDS, 16 bytes per lane, tracked by ASYNCcnt.
__device__ __forceinline__ void async_ld128(unsigned lds_off, const float* gptr) {
    asm volatile("global_load_async_to_lds_b128 %0, %1, off"
                 :: "v"(lds_off), "v"(gptr)
                 : "memory");
}

// Each workgroup computes the full 40x128 correlation slice for one (b,h).
// out[w, w'] = sum_c L[c,w] * R[c,w'], band d = w - w' in [0,40).
__global__ __launch_bounds__(256)
void dispnetc_corr_kernel(const float* __restrict__ gL,
                          const float* __restrict__ gR,
                          float* __restrict__ gOut) {
    __shared__ __align__(16) float sL0[CHUNK * LPITCH];   // 8704 B each
    __shared__ __align__(16) float sL1[CHUNK * LPITCH];
    __shared__ __align__(16) float sR0[CHUNK * LPITCH];
    __shared__ __align__(16) float sR1[CHUNK * LPITCH];
    __shared__ float sOut[D_MAX * OPITCH];                // 20640 B (total 55456 B)

    const int bh   = blockIdx.x;
    const int b    = bh >> 6;        // / 64
    const int h    = bh & 63;
    const int tid  = threadIdx.x;
    const int lane = tid & 31;
    const int half = lane >> 4;      // 0: lanes 0-15, 1: lanes 16-31
    const int l15  = lane & 15;

    // Wave id is uniform: pin it to an SGPR so all tile bookkeeping is scalar
    // (avoids saveexec/EXEC-masking around the WMMA blocks).
    const int wave = __builtin_amdgcn_readfirstlane(tid >> 5);   // 0..7

    // Zero the output staging buffer (supplies the w<d zero padding).
    for (int f = tid; f < D_MAX * OPITCH; f += 256) sOut[f] = 0.0f;

    // ---- Band tile list: 26 tiles (i = w-tile 0..7, delta = i-j in 0..min(i,3)).
    int ti[4], tj[4];
    int ntiles = 0;
#pragma unroll
    for (int u = 0; u < 4; ++u) {
        const int t = wave + 8 * u;                       // scalar
        int i = 0, base = 0;
        if (t < 1)      { i = 0; base = 0; }
        else if (t < 3) { i = 1; base = 1; }
        else if (t < 6) { i = 2; base = 3; }
        else            { i = 3 + (t - 6) / 4; base = 6 + ((t - 6) / 4) * 4; }
        const int delta = t - base;
        ti[u] = i;
        tj[u] = i - delta;
        if (t < 26) ntiles = u + 1;
    }
    ntiles = __builtin_amdgcn_readfirstlane(ntiles);      // scalar branch condition

    // Per-tile lane columns, invariant across the K loop.
    int ibx[4], jbx[4];
#pragma unroll
    for (int u = 0; u < 4; ++u) {
        ibx[u] = 16 * ti[u] + l15;   // A lane column (w)
        jbx[u] = 16 * tj[u] + l15;   // B lane column (w')
    }

    v8f acc0 = {}, acc1 = {}, acc2 = {}, acc3 = {};

    const float* __restrict__ pL = gL + (size_t)b * CHW + (size_t)h * WW;
    const float* __restrict__ pR = gR + (size_t)b * CHW + (size_t)h * WW;

    // ---- Per-thread async-load descriptors: 2 x b128 per matrix per chunk.
    // f = tid + q*256 over [0,512): ch = f>>5 (0..15), c4 = (f&31)*4 floats.
    const int f0  = tid;
    const int f1  = tid + 256;
    const int ch0 = f0 >> 5,           ch1 = f1 >> 5;
    const int c40 = (f0 & 31) << 2,    c41 = (f1 & 31) << 2;
    const unsigned o0 = (unsigned)(ch0 * LPITCH + c40) * 4u;   // LDS byte offset in buffer
    const unsigned o1 = (unsigned)(ch1 * LPITCH + c41) * 4u;
    const size_t   g0 = (size_t)ch0 * HW + c40;                // float offset in chunk
    const size_t   g1 = (size_t)ch1 * HW + c41;

    // LDS base byte offsets (flat LDS addresses carry the offset in addr[31:0]).
    const unsigned bL0 = (unsigned)(uintptr_t)&sL0[0];
    const unsigned bL1 = (unsigned)(uintptr_t)&sL1[0];
    const unsigned bR0 = (unsigned)(uintptr_t)&sR0[0];
    const unsigned bR1 = (unsigned)(uintptr_t)&sR1[0];

    const int ro = (2 * half) * LPITCH;   // lanes 16-31 hold K=2,3 of each K4 group

    // Prologue: kick off chunk 0 into buffer 0.
    async_ld128(bL0 + o0, pL + g0);
    async_ld128(bL0 + o1, pL + g1);
    async_ld128(bR0 + o0, pR + g0);
    async_ld128(bR0 + o1, pR + g1);

    for (int k = 0; k < NCHUNK; ++k) {
        const int p = k & 1;

        // All waves done reading buf[1-p] (chunk k-1) before DMA overwrites it.
        __syncthreads();

        if (k + 1 < NCHUNK) {
            const size_t cb = (size_t)(k + 1) * CHUNK * HW;
            const float* lp = pL + cb;
            const float* rp = pR + cb;
            const unsigned lb = p ? bL0 : bL1;   // next buffer = 1-p
            const unsigned rb = p ? bR0 : bR1;
            async_ld128(lb + o0, lp + g0);
            async_ld128(lb + o1, lp + g1);
            async_ld128(rb + o0, rp + g0);
            async_ld128(rb + o1, rp + g1);
            // 8 outstanding; <=4 left  =>  chunk k's four loads have landed.
            asm volatile("s_wait_asynccnt 0x4" ::: "memory");
        } else {
            asm volatile("s_wait_asynccnt 0x0" ::: "memory");
        }

        // Chunk k visible to the whole workgroup.
        __syncthreads();

        const float* sLp = p ? sL1 : sL0;
        const float* sRp = p ? sR1 : sR0;

        // WMMA over this K-chunk for each owned band tile (chunk k+1 DMA overlaps).
#pragma unroll
        for (int u = 0; u < 4; ++u) {
            if (u >= ntiles) break;              // scalar, wave-uniform
            const int ib = ibx[u];
            const int jb = jbx[u];
            v8f c = (u == 0) ? acc0 : (u == 1) ? acc1 : (u == 2) ? acc2 : acc3;
#pragma unroll
            for (int k4 = 0; k4 < CHUNK / 4; ++k4) {
                const int r0 = ro + 4 * k4 * LPITCH;
                v2f a, bb;
                a.x  = sLp[r0 + ib];
                a.y  = sLp[r0 + LPITCH + ib];
                bb.x = sRp[r0 + jb];
                bb.y = sRp[r0 + LPITCH + jb];
                c = __builtin_amdgcn_wmma_f32_16x16x4_f32(
                        /*neg_a=*/false, a, /*neg_b=*/false, bb,
                        /*c_mod=*/(short)0, c,
                        /*reuse_a=*/false, /*reuse_b=*/false);
            }
            if (u == 0) acc0 = c; else if (u == 1) acc1 = c;
            else if (u == 2) acc2 = c; else acc3 = c;
        }
    }

    __syncthreads();

    // Scatter accumulators into the banded [d][w] staging buffer (apply 1/C).
    const float inv_c = 1.0f / (float)C_TOT;
#pragma unroll
    for (int u = 0; u < 4; ++u) {
        if (u >= ntiles) break;      // scalar, wave-uniform
        const v8f c = (u == 0) ? acc0 : (u == 1) ? acc1 : (u == 2) ? acc2 : acc3;
        const int wbase  = 16 * ti[u];
        const int wpbase = 16 * tj[u];
#pragma unroll
        for (int r = 0; r < 8; ++r) {
            const int M  = r + 8 * half;          // D row (w offset)
            const int w  = wbase + M;
            const int wp = wpbase + l15;          // D col (w' offset)
            const int d  = w - wp;
            if (d >= 0 && d < D_MAX) sOut[d * OPITCH + w] = c[r] * inv_c;
        }
    }
    __syncthreads();

    // Coalesced store of the full 40x128 slice: out[((b*40+d)*64+h)*128 + w].
    float* __restrict__ pO = gOut + ((size_t)b * D_MAX * HH + h) * WW;
    for (int f = tid; f < D_MAX * WW; f += 256) {
        const int d = f >> 7;
        const int w = f & 127;
        pO[(size_t)d * HW + w] = sOut[d * OPITCH + w];
    }
}

extern "C" void kernel_launch(void* const* d_in, const int* in_sizes, int n_in,
                              void* d_out, int out_size, void* d_ws, size_t ws_size,
                              hipStream_t stream) {
    (void)in_sizes; (void)n_in; (void)out_size; (void)d_ws; (void)ws_size;
    const float* conv3a_l = (const float*)d_in[0];
    const float* conv3a_r = (const float*)d_in[1];
    float* out = (float*)d_out;
    // One workgroup per (b,h): 8 * 64 = 512 workgroups, 256 threads each.
    dispnetc_corr_kernel<<<dim3(512), dim3(256), 0, stream>>>(conv3a_l, conv3a_r, out);
}